// GCN_11312943857819
// MI455X (gfx1250) — compile-verified
//
#include <hip/hip_runtime.h>

typedef float v2f __attribute__((ext_vector_type(2)));
typedef float v8f __attribute__((ext_vector_type(8)));

#define D_IN  512
#define H1    16
#define D_OUT 64

// ---------------- normalization pipeline ----------------

__global__ void k_init_deg(float* __restrict__ deg, int N) {
    int i = blockIdx.x * blockDim.x + threadIdx.x;
    if (i < N) deg[i] = 1.0f;   // self-loop weight contribution
}

__global__ void k_deg_acc(const int* __restrict__ ei, const float* __restrict__ w,
                          float* __restrict__ deg, int E) {
    int e = blockIdx.x * blockDim.x + threadIdx.x;
    if (e < E) atomicAdd(&deg[ei[E + e]], w[e]);   // col = dst
}

__global__ void k_dinv(float* __restrict__ deg, int N) {
    int i = blockIdx.x * blockDim.x + threadIdx.x;
    if (i < N) {
        float d = deg[i];
        deg[i] = (d > 0.0f) ? rsqrtf(d) : 0.0f;
    }
}

__global__ void k_norm(const int* __restrict__ ei, const float* __restrict__ w,
                       const float* __restrict__ dinv, float* __restrict__ nrm, int E) {
    int e = blockIdx.x * blockDim.x + threadIdx.x;
    if (e < E) nrm[e] = dinv[ei[e]] * w[e] * dinv[ei[E + e]];
}

// ---------------- GEMM1: h1 = x @ W1  ([N,512] x [512,16]) via V_WMMA_F32_16X16X4_F32 ----

__global__ __launch_bounds__(256) void k_gemm1(const float* __restrict__ x,
                                               const float* __restrict__ W1,
                                               float* __restrict__ h1, int N) {
    const int lane = threadIdx.x & 31;
    const int wave = threadIdx.x >> 5;          // 8 waves per block (wave32)
    const int tile = blockIdx.x * 8 + wave;     // 16 rows per tile
    const int ntiles = (N + 15) >> 4;
    if (tile >= ntiles) return;                 // wave-uniform: EXEC stays all-ones

    const int row0 = tile * 16;
    const int col  = lane & 15;                 // output column / B column / A row
    const int kofs = (lane >> 4) << 1;          // 0 or 2 (K split across lane halves)

    int r = row0 + col; if (r > N - 1) r = N - 1;
    const float* xrow = x + (size_t)r * D_IN;
    const float* bcol = W1 + col;

    v8f acc = {};
    for (int k = 0; k < D_IN; k += 4) {
        v2f a = *(const v2f*)(xrow + k + kofs);          // A[M=lane&15][K=k+kofs, +1]
        v2f b;
        b[0] = bcol[(size_t)(k + kofs)     * H1];        // B[K][N=col]
        b[1] = bcol[(size_t)(k + kofs + 1) * H1];
        acc = __builtin_amdgcn_wmma_f32_16x16x4_f32(
                  false, a, false, b, (short)0, acc, false, false);
    }

    const int mhi = (lane >> 4) << 3;           // lanes 16-31 hold M = v+8
    if (row0 + 16 <= N) {                       // full tile: unguarded stores (common path)
#pragma unroll
        for (int v = 0; v < 8; ++v)
            h1[(size_t)(row0 + v + mhi) * H1 + col] = acc[v];
    } else {
#pragma unroll
        for (int v = 0; v < 8; ++v) {
            int rr = row0 + v + mhi;
            if (rr < N) h1[(size_t)rr * H1 + col] = acc[v];
        }
    }
}

// seed aggregation with self-loop message + bias:  h1agg[n,f] = h1[n,f]*dinv[n]^2 + b1[f]
// float4 over the 16-wide feature dim: 4 threads per node
__global__ void k_seed1(const float* __restrict__ h1, const float* __restrict__ dinv,
                        const float* __restrict__ b1, float* __restrict__ h1agg, int N) {
    int t = blockIdx.x * blockDim.x + threadIdx.x;      // N*4 threads
    if (t < N * (H1 / 4)) {
        int n = t >> 2, q = (t & 3) * 4;
        float di = dinv[n];
        float d2 = di * di;
        float4 hv = *(const float4*)(h1 + (size_t)n * H1 + q);
        float4 bv = *(const float4*)(b1 + q);
        float4 o;
        o.x = hv.x * d2 + bv.x; o.y = hv.y * d2 + bv.y;
        o.z = hv.z * d2 + bv.z; o.w = hv.w * d2 + bv.w;
        *(float4*)(h1agg + (size_t)n * H1 + q) = o;
    }
}

// edge scatter layer 1: 4 threads per edge, float4 gather + 4 atomics
__global__ void k_prop1(const int* __restrict__ ei, const float* __restrict__ nrm,
                        const float* __restrict__ h1, float* __restrict__ h1agg, int E) {
    int t = blockIdx.x * blockDim.x + threadIdx.x;      // E*4 threads
    if (t < E * (H1 / 4)) {
        int e = t >> 2, q = (t & 3) * 4;
        int s = ei[e], d = ei[E + e];
        float nv = nrm[e];
        float4 hv = *(const float4*)(h1 + (size_t)s * H1 + q);
        float* o = h1agg + (size_t)d * H1 + q;
        atomicAdd(o + 0, hv.x * nv);
        atomicAdd(o + 1, hv.y * nv);
        atomicAdd(o + 2, hv.z * nv);
        atomicAdd(o + 3, hv.w * nv);
    }
}

// ---------------- GEMM2: hw2 = relu(h1agg) @ W2  ([N,16] x [16,64]) --------------------

__global__ __launch_bounds__(256) void k_gemm2(const float* __restrict__ h,
                                               const float* __restrict__ W2,
                                               float* __restrict__ hw2, int N) {
    const int lane = threadIdx.x & 31;
    const int wave = threadIdx.x >> 5;
    const int tile = blockIdx.x * 8 + wave;
    const int ntiles = (N + 15) >> 4;
    if (tile >= ntiles) return;

    const int row0 = tile * 16;
    const int col  = lane & 15;
    const int kofs = (lane >> 4) << 1;

    int r = row0 + col; if (r > N - 1) r = N - 1;
    const float* hrow = h + (size_t)r * H1;

    v8f acc0 = {}, acc1 = {}, acc2 = {}, acc3 = {};
#pragma unroll
    for (int k = 0; k < H1; k += 4) {
        v2f a;
        a[0] = fmaxf(hrow[k + kofs],     0.0f);   // fused ReLU on A-fragment load
        a[1] = fmaxf(hrow[k + kofs + 1], 0.0f);
        const float* b0 = W2 + (size_t)(k + kofs)     * D_OUT + col;
        const float* b1 = W2 + (size_t)(k + kofs + 1) * D_OUT + col;
        v2f b;
        b[0] = b0[0];  b[1] = b1[0];
        acc0 = __builtin_amdgcn_wmma_f32_16x16x4_f32(false, a, false, b, (short)0, acc0, false, false);
        b[0] = b0[16]; b[1] = b1[16];
        acc1 = __builtin_amdgcn_wmma_f32_16x16x4_f32(false, a, false, b, (short)0, acc1, false, false);
        b[0] = b0[32]; b[1] = b1[32];
        acc2 = __builtin_amdgcn_wmma_f32_16x16x4_f32(false, a, false, b, (short)0, acc2, false, false);
        b[0] = b0[48]; b[1] = b1[48];
        acc3 = __builtin_amdgcn_wmma_f32_16x16x4_f32(false, a, false, b, (short)0, acc3, false, false);
    }

    const int mhi = (lane >> 4) << 3;
    if (row0 + 16 <= N) {
#pragma unroll
        for (int v = 0; v < 8; ++v) {
            float* o = hw2 + (size_t)(row0 + v + mhi) * D_OUT + col;
            o[0]  = acc0[v];
            o[16] = acc1[v];
            o[32] = acc2[v];
            o[48] = acc3[v];
        }
    } else {
#pragma unroll
        for (int v = 0; v < 8; ++v) {
            int rr = row0 + v + mhi;
            if (rr < N) {
                float* o = hw2 + (size_t)rr * D_OUT + col;
                o[0]  = acc0[v];
                o[16] = acc1[v];
                o[32] = acc2[v];
                o[48] = acc3[v];
            }
        }
    }
}

// seed output with self-loop message + bias: out[n,f] = hw2[n,f]*dinv[n]^2 + b2[f]
// float4 over 64-wide feature dim: 16 threads per node
__global__ void k_seed2(const float* __restrict__ hw2, const float* __restrict__ dinv,
                        const float* __restrict__ b2, float* __restrict__ out, int N) {
    int t = blockIdx.x * blockDim.x + threadIdx.x;      // N*16 threads
    if (t < N * (D_OUT / 4)) {
        int n = t >> 4, q = (t & 15) * 4;
        float di = dinv[n];
        float d2 = di * di;
        float4 hv = *(const float4*)(hw2 + (size_t)n * D_OUT + q);
        float4 bv = *(const float4*)(b2 + q);
        float4 o;
        o.x = hv.x * d2 + bv.x; o.y = hv.y * d2 + bv.y;
        o.z = hv.z * d2 + bv.z; o.w = hv.w * d2 + bv.w;
        *(float4*)(out + (size_t)n * D_OUT + q) = o;
    }
}

// edge scatter layer 2: 16 threads per edge, float4 gather + 4 atomics
__global__ void k_prop2(const int* __restrict__ ei, const float* __restrict__ nrm,
                        const float* __restrict__ hw2, float* __restrict__ out, int E) {
    int t = blockIdx.x * blockDim.x + threadIdx.x;      // E*16 threads
    if (t < E * (D_OUT / 4)) {
        int e = t >> 4, q = (t & 15) * 4;
        int s = ei[e], d = ei[E + e];
        float nv = nrm[e];
        float4 hv = *(const float4*)(hw2 + (size_t)s * D_OUT + q);
        float* o = out + (size_t)d * D_OUT + q;
        atomicAdd(o + 0, hv.x * nv);
        atomicAdd(o + 1, hv.y * nv);
        atomicAdd(o + 2, hv.z * nv);
        atomicAdd(o + 3, hv.w * nv);
    }
}

// ---------------- launcher ----------------

extern "C" void kernel_launch(void* const* d_in, const int* in_sizes, int n_in,
                              void* d_out, int out_size, void* d_ws, size_t ws_size,
                              hipStream_t stream) {
    const float* x  = (const float*)d_in[0];
    const int*   ei = (const int*)  d_in[1];
    const float* w  = (const float*)d_in[2];
    const float* W1 = (const float*)d_in[3];
    const float* b1 = (const float*)d_in[4];
    const float* W2 = (const float*)d_in[5];
    const float* b2 = (const float*)d_in[6];
    float* out = (float*)d_out;

    const int N = in_sizes[0] / D_IN;     // 100000
    const int E = in_sizes[2];            // 3200000

    // workspace layout (floats): deg/dinv[N] | norm[E] | h1[N*16] | h1agg[N*16] | hw2[N*64]
    // all offsets are multiples of 4 floats -> float4 accesses stay 16B-aligned
    float* deg   = (float*)d_ws;
    float* nrm   = deg + N;
    float* h1    = nrm + E;
    float* h1agg = h1 + (size_t)N * H1;
    float* hw2   = h1agg + (size_t)N * H1;

    const int T = 256;
    const int ntiles = (N + 15) >> 4;
    const int gemmBlocks = (ntiles + 7) / 8;

    k_init_deg<<<(N + T - 1) / T, T, 0, stream>>>(deg, N);
    k_deg_acc <<<(E + T - 1) / T, T, 0, stream>>>(ei, w, deg, E);
    k_dinv    <<<(N + T - 1) / T, T, 0, stream>>>(deg, N);
    k_norm    <<<(E + T - 1) / T, T, 0, stream>>>(ei, w, deg, nrm, E);

    k_gemm1   <<<gemmBlocks, T, 0, stream>>>(x, W1, h1, N);
    k_seed1   <<<((N * 4)  + T - 1) / T, T, 0, stream>>>(h1, deg, b1, h1agg, N);
    k_prop1   <<<((E * 4)  + T - 1) / T, T, 0, stream>>>(ei, nrm, h1, h1agg, E);

    k_gemm2   <<<gemmBlocks, T, 0, stream>>>(h1agg, W2, hw2, N);
    k_seed2   <<<((N * 16) + T - 1) / T, T, 0, stream>>>(hw2, deg, b2, out, N);
    k_prop2   <<<((E * 16) + T - 1) / T, T, 0, stream>>>(ei, nrm, hw2, out, E);
}